// HarmonicBondTorchForce_26207890440209
// MI455X (gfx1250) — compile-verified
//
#include <hip/hip_runtime.h>

typedef float v2f __attribute__((ext_vector_type(2)));
typedef float v8f __attribute__((ext_vector_type(8)));
typedef int   v2i __attribute__((ext_vector_type(2)));

#define TPB   256   // 8 wave32 waves per block
#define ITEMS 16    // bonds per thread -> 4096 bonds per block

// Pass 1: each block reduces a contiguous 4096-bond tile to one partial sum.
// Streams (bonds, b0, k) use non-temporal loads (TH=NT) so 64MB of single-use
// stream data doesn't evict the 48MB coords working set from the 192MB L2;
// the random coord gathers then run at L2 bandwidth, leaving HBM (23.3 TB/s)
// for the streams -> ~5-7us roofline for 4M bonds.
__global__ __launch_bounds__(TPB) void bond_energy_partial(
    const float* __restrict__ coords,
    const v2i*   __restrict__ bonds,
    const float* __restrict__ b0,
    const float* __restrict__ kf,
    float*       __restrict__ partials,
    int nBonds)
{
    const int tid = threadIdx.x;
    const long long base = (long long)blockIdx.x * (TPB * ITEMS);

    float e = 0.0f;
#pragma unroll 4
    for (int j = 0; j < ITEMS; ++j) {
        long long idx = base + (long long)j * TPB + tid;
        if (idx < (long long)nBonds) {
            v2i   bnd = __builtin_nontemporal_load(&bonds[idx]);  // b64 NT stream
            float r0  = __builtin_nontemporal_load(&b0[idx]);     // NT stream
            float kk  = __builtin_nontemporal_load(&kf[idx]);     // NT stream
            const float* __restrict__ pi = coords + 3ll * (long long)bnd.x;
            const float* __restrict__ pj = coords + 3ll * (long long)bnd.y;
            float dx = pi[0] - pj[0];
            float dy = pi[1] - pj[1];
            float dz = pi[2] - pj[2];
            float r2 = fmaf(dx, dx, fmaf(dy, dy, dz * dz));
            float r  = sqrtf(r2);
            float d  = r - r0;
            e = fmaf(0.5f * kk * d, d, e);
        }
    }

    // wave32 butterfly reduction (CDNA5 is wave32: 5 steps, width 32)
    for (int off = 16; off > 0; off >>= 1)
        e += __shfl_xor(e, off, 32);

    __shared__ float wsum[TPB / 32];
    if ((tid & 31) == 0) wsum[tid >> 5] = e;
    __syncthreads();
    if (tid == 0) {
        float s = 0.0f;
#pragma unroll
        for (int w = 0; w < TPB / 32; ++w) s += wsum[w];
        partials[blockIdx.x] = s;   // deterministic: fixed order, no atomics
    }
}

// Pass 2: one 256-thread block. The ~1K partials are loaded by 8 waves in
// parallel (~4 strided iterations instead of 16 serialized L2 round-trips on
// one wave), staged through LDS, then folded into the 64 slots of a 16x4 f32
// A-matrix. One V_WMMA_F32_16X16X4_F32 against an all-ones B matrix performs
// the cross-lane sum: D[m][n] = sum_k A[m][k]; D VGPRs 0..7 hold rows 0..7 on
// lanes 0-15 and rows 8..15 on lanes 16-31, so total = s + shfl_xor(s,16).
// ALL waves execute the identical WMMA on the same LDS-broadcast data (no
// divergent branch around it -> EXEC is all-ones, as WMMA requires); the
// result is wave-uniform and only thread 0 stores it.
__global__ __launch_bounds__(TPB) void final_reduce(
    const float* __restrict__ partials, int n, float* __restrict__ out)
{
    __shared__ float lds[TPB];
    const int tid  = threadIdx.x;
    const int lane = tid & 31;

    float a = 0.0f;
    for (int i = tid; i < n; i += TPB)     // parallel strided loads, 8 waves
        a += partials[i];
    lds[tid] = a;
    __syncthreads();

    // Fold 256 lane-sums into the 64 A-matrix slots (2 VGPRs x 32 lanes).
    float a0 = lds[lane]      + lds[lane + 64]  + lds[lane + 128] + lds[lane + 192];
    float a1 = lds[lane + 32] + lds[lane + 96]  + lds[lane + 160] + lds[lane + 224];

#if __has_builtin(__builtin_amdgcn_wmma_f32_16x16x4_f32)
    v2f A; A.x = a0;   A.y = a1;     // A 16x4 f32 operand
    v2f B; B.x = 1.0f; B.y = 1.0f;   // all-ones B: layout-independent
    v8f C = {};
    v8f D = __builtin_amdgcn_wmma_f32_16x16x4_f32(
        /*neg_a=*/false, A, /*neg_b=*/false, B,
        /*c_mod=*/(short)0, C, /*reuse_a=*/false, /*reuse_b=*/false);
    float s = D[0] + D[1] + D[2] + D[3] + D[4] + D[5] + D[6] + D[7];
    s += __shfl_xor(s, 16, 32);      // rows 0-7 half + rows 8-15 half
#else
    float s = a0 + a1;
    for (int off = 16; off > 0; off >>= 1)
        s += __shfl_xor(s, off, 32);
#endif

    if (tid == 0) out[0] = s;
}

extern "C" void kernel_launch(void* const* d_in, const int* in_sizes, int n_in,
                              void* d_out, int out_size, void* d_ws, size_t ws_size,
                              hipStream_t stream) {
    // inputs: 0=coords(f32 Nx3) 1=box(unused) 2=bonds(i32 Nx2) 3=b0(f32) 4=k(f32)
    const float* coords = (const float*)d_in[0];
    const v2i*   bonds  = (const v2i*)  d_in[2];
    const float* b0     = (const float*)d_in[3];
    const float* kf     = (const float*)d_in[4];
    const int nBonds = in_sizes[3];

    const int tile = TPB * ITEMS;
    const int nb   = (nBonds + tile - 1) / tile;   // ~977 blocks for 4M bonds

    float* partials = (float*)d_ws;                // nb * 4 bytes of scratch

    bond_energy_partial<<<nb, TPB, 0, stream>>>(coords, bonds, b0, kf,
                                                partials, nBonds);
    final_reduce<<<1, TPB, 0, stream>>>(partials, nb, (float*)d_out);
}